// CrossLocalAttentionLayer_32246614458840
// MI455X (gfx1250) — compile-verified
//
#include <hip/hip_runtime.h>
#include <hip/hip_bf16.h>

// Problem constants (B=1)
#define NN   25000     // nodes
#define EE   400000    // edges
#define DD   128       // feature dim (= H*DK = H*DV)
#define HH   4         // heads
#define DKV  32        // per-head dim
#define LS   132       // LDS row stride (128 + 4 pad -> no bank conflicts)

typedef __attribute__((ext_vector_type(2))) float v2f;
typedef __attribute__((ext_vector_type(8))) float v8f;

// D = A(16x4,f32) * B(4x16,f32) + C(16x16,f32)  — full-precision CDNA5 WMMA
__device__ __forceinline__ v8f wmma4f32(v2f a, v2f b, v8f c) {
  return __builtin_amdgcn_wmma_f32_16x16x4_f32(
      /*neg_a=*/false, a, /*neg_b=*/false, b,
      /*c_mod=*/(short)0, c, /*reuse_a=*/false, /*reuse_b=*/false);
}

// Per-wave GEMM: C(16x32) = ldsX(16x128) * W^T, W row-major [128 out][128 in].
// wave w covers output cols n0 = 32*w .. 32*w+31 as two 16-wide WMMA tiles.
__device__ __forceinline__ void wave_gemm(const float* __restrict__ ldsX,
                                          const float* __restrict__ W,
                                          int lane, int wave,
                                          v8f& c0, v8f& c1) {
  const int hi = lane >> 4;       // 0: K pair {0,1}, 1: K pair {2,3}
  const int lm = lane & 15;       // M (for A) / N (for B)
  const int n0 = wave * 32;
  c0 = v8f{}; c1 = v8f{};
#pragma unroll
  for (int k0 = 0; k0 < DD; k0 += 4) {
    v2f a;
    a.x = ldsX[lm * LS + k0 + 2 * hi];
    a.y = ldsX[lm * LS + k0 + 2 * hi + 1];
    const float* wr0 = W + (size_t)(n0 + lm) * DD + k0 + 2 * hi;        // B[k][n]=W[n][k]
    const float* wr1 = W + (size_t)(n0 + 16 + lm) * DD + k0 + 2 * hi;
    v2f b0; b0.x = wr0[0]; b0.y = wr0[1];
    v2f b1; b1.x = wr1[0]; b1.y = wr1[1];
    c0 = wmma4f32(a, b0, c0);
    c1 = wmma4f32(a, b1, c1);
  }
}

// Cooperative load of a 16x128 fp32 tile into padded LDS (zero-fill OOB rows).
__device__ __forceinline__ void stage_tile(float* __restrict__ lds,
                                           const float* __restrict__ src,
                                           int row0, int nrows, int tid) {
#pragma unroll
  for (int i = 0; i < 4; ++i) {
    int idx = tid + i * 128;            // float4 index in [0,512)
    int row = idx >> 5;                 // 32 float4 per row
    int c4  = idx & 31;
    float4 v = make_float4(0.f, 0.f, 0.f, 0.f);
    int gr = row0 + row;
    if (gr < nrows) v = *(const float4*)(src + (size_t)gr * DD + c4 * 4);
    *(float4*)&lds[row * LS + c4 * 4] = v;
  }
}

// ---------------- Kernel 1: projection  Y[n,:] = X[n,:] @ W^T ----------------
__global__ __launch_bounds__(128) void proj_kernel(const float* __restrict__ X,
                                                   const float* __restrict__ W,
                                                   float* __restrict__ Y,
                                                   int nrows) {
  __shared__ __align__(16) float ldsX[16 * LS];
  const int tid = threadIdx.x;
  const int m0  = blockIdx.x * 16;
  stage_tile(ldsX, X, m0, nrows, tid);
  __syncthreads();

  const int lane = tid & 31, wave = tid >> 5;
  const int hi = lane >> 4, lm = lane & 15, n0 = wave * 32;
  v8f c0, c1;
  wave_gemm(ldsX, W, lane, wave, c0, c1);
#pragma unroll
  for (int r = 0; r < 8; ++r) {
    int m = m0 + r + 8 * hi;
    if (m < nrows) {
      Y[(size_t)m * DD + n0 + lm]      = c0[r];
      Y[(size_t)m * DD + n0 + 16 + lm] = c1[r];
    }
  }
}

// ------------- Kernel 2: edge phase with fused Ef projection -----------------
__global__ __launch_bounds__(128) void edge_kernel(const int*   __restrict__ eidx,
                                                   const float* __restrict__ EF,
                                                   const float* __restrict__ WE,
                                                   const float* __restrict__ Q,
                                                   const float* __restrict__ K,
                                                   const float* __restrict__ V,
                                                   float* __restrict__ OUTACC,
                                                   float* __restrict__ COEFF,
                                                   float* __restrict__ ATTN) {
  __shared__ __align__(16) float ldsX[16 * LS];   // edge_features tile
  __shared__ __align__(16) float ldsE[16 * LS];   // projected Ef tile
  const int tid = threadIdx.x;
  const int e0  = blockIdx.x * 16;
  stage_tile(ldsX, EF, e0, EE, tid);
  __syncthreads();

  const int lane = tid & 31, wave = tid >> 5;
  const int hi = lane >> 4, lm = lane & 15, n0 = wave * 32;
  v8f c0, c1;
  wave_gemm(ldsX, WE, lane, wave, c0, c1);        // Ef tile = Xe @ WE^T
#pragma unroll
  for (int r = 0; r < 8; ++r) {
    ldsE[(r + 8 * hi) * LS + n0 + lm]      = c0[r];
    ldsE[(r + 8 * hi) * LS + n0 + 16 + lm] = c1[r];
  }
  __syncthreads();

  if (tid < 64) {                                  // one (edge, head) per thread
    const int el = tid >> 2, h = tid & 3;
    const int e  = e0 + el;
    if (e < EE) {
      const int src = eidx[e];
      const int tgt = eidx[EE + e];
      const float* qr = Q + (size_t)src * DD + h * DKV;
      const float* kr = K + (size_t)tgt * DD + h * DKV;
      const float* vr = V + (size_t)tgt * DD + h * DKV;
      const float* er = &ldsE[el * LS + h * DKV];
      float s = 0.f;
#pragma unroll
      for (int d = 0; d < DKV; ++d) s += qr[d] * kr[d] * er[d];
      s *= 0.17677669529663687f;                   // 1/sqrt(32)
      s = fminf(5.f, fmaxf(-5.f, s));
      const float attn = expf(s);
      ATTN[(size_t)h * EE + e] = attn;             // attn output, flat (H,E)
      float* oacc = OUTACC + (size_t)src * DD + h * DKV;
#pragma unroll
      for (int d = 0; d < DKV; ++d) atomicAdd(&oacc[d], attn * vr[d]);
      atomicAdd(&COEFF[(size_t)src * HH + h], attn);
    }
  }
}

// ------- Kernel 3: normalize, out @ W_fc^T, residual, LayerNorm --------------
__global__ __launch_bounds__(128) void final_kernel(const float* __restrict__ OUTACC,
                                                    const float* __restrict__ COEFF,
                                                    const float* __restrict__ Wfc,
                                                    const float* __restrict__ inQ,
                                                    const float* __restrict__ gamma,
                                                    const float* __restrict__ beta,
                                                    float* __restrict__ Y) {
  __shared__ __align__(16) float ldsX[16 * LS];
  __shared__ __align__(16) float ldsY[16 * LS];
  const int tid = threadIdx.x;
  const int m0  = blockIdx.x * 16;

  // stage OUTACC / (coeff + 1e-8)
#pragma unroll
  for (int i = 0; i < 4; ++i) {
    int idx = tid + i * 128;
    int row = idx >> 5;
    int c4  = idx & 31;
    int n   = m0 + row;
    float4 v = make_float4(0.f, 0.f, 0.f, 0.f);
    if (n < NN) {
      v = *(const float4*)(OUTACC + (size_t)n * DD + c4 * 4);
      int h = (c4 * 4) >> 5;
      float inv = 1.f / (COEFF[(size_t)n * HH + h] + 1e-8f);
      v.x *= inv; v.y *= inv; v.z *= inv; v.w *= inv;
    }
    *(float4*)&ldsX[row * LS + c4 * 4] = v;
  }
  __syncthreads();

  const int lane = tid & 31, wave = tid >> 5;
  const int hi = lane >> 4, lm = lane & 15, n0 = wave * 32;
  v8f c0, c1;
  wave_gemm(ldsX, Wfc, lane, wave, c0, c1);
#pragma unroll
  for (int r = 0; r < 8; ++r) {
    ldsY[(r + 8 * hi) * LS + n0 + lm]      = c0[r];
    ldsY[(r + 8 * hi) * LS + n0 + 16 + lm] = c1[r];
  }
  __syncthreads();

  if (tid < 16) {
    const int n = m0 + tid;
    if (n < NN) {
      const float* xin = inQ + (size_t)n * DD;
      float* row = &ldsY[tid * LS];
      float s1 = 0.f, s2 = 0.f;
      for (int c = 0; c < DD; ++c) {
        float x = row[c] + xin[c];                 // residual
        row[c] = x;
        s1 += x; s2 += x * x;
      }
      const float mu  = s1 * (1.f / DD);
      const float var = s2 * (1.f / DD) - mu * mu;
      const float rst = rsqrtf(var + 1e-5f);
      for (int c = 0; c < DD; ++c)
        Y[(size_t)n * DD + c] = (row[c] - mu) * rst * gamma[c] + beta[c];
    }
  }
}

extern "C" void kernel_launch(void* const* d_in, const int* in_sizes, int n_in,
                              void* d_out, int out_size, void* d_ws, size_t ws_size,
                              hipStream_t stream) {
  (void)in_sizes; (void)n_in; (void)out_size; (void)ws_size;
  const int*   eidx  = (const int*)d_in[0];     // (1,2,E)
  const float* EF    = (const float*)d_in[1];   // (E,128)
  const float* inQ   = (const float*)d_in[2];
  const float* inK   = (const float*)d_in[3];
  const float* inV   = (const float*)d_in[4];
  const float* WQ    = (const float*)d_in[5];
  const float* WK    = (const float*)d_in[6];
  const float* WV    = (const float*)d_in[7];
  const float* WE    = (const float*)d_in[8];
  const float* Wfc   = (const float*)d_in[9];
  const float* gamma = (const float*)d_in[10];
  const float* beta  = (const float*)d_in[11];

  float* ws     = (float*)d_ws;                 // ~52 MB total
  float* Q      = ws;
  float* K      = Q + (size_t)NN * DD;
  float* V      = K + (size_t)NN * DD;
  float* OUTACC = V + (size_t)NN * DD;
  float* COEFF  = OUTACC + (size_t)NN * DD;

  float* y    = (float*)d_out;                  // (N,128)
  float* attn = y + (size_t)NN * DD;            // (H,E) flat

  hipMemsetAsync(OUTACC, 0, ((size_t)NN * DD + (size_t)NN * HH) * sizeof(float), stream);

  const int mt = (NN + 15) / 16;                // 1563 node tiles
  proj_kernel<<<mt, 128, 0, stream>>>(inQ, WQ, Q, NN);
  proj_kernel<<<mt, 128, 0, stream>>>(inK, WK, K, NN);
  proj_kernel<<<mt, 128, 0, stream>>>(inV, WV, V, NN);
  edge_kernel<<<EE / 16, 128, 0, stream>>>(eidx, EF, WE, Q, K, V, OUTACC, COEFF, attn);
  final_kernel<<<mt, 128, 0, stream>>>(OUTACC, COEFF, Wfc, inQ, gamma, beta, y);
}